// KnowledgeLayer_53274774340198
// MI455X (gfx1250) — compile-verified
//
#include <hip/hip_runtime.h>
#include <cmath>

// ---------------- problem geometry (fixed by the reference) ----------------
#define NVARS   1024          // pos rows
#define BATCH   4096          // batch columns
#define N0      8192
#define N1      4096
#define N2      8192
#define N3      2048
#define BC      4             // batch columns per workgroup (LDS tile width)
#define THREADS 512           // 16 wave32
#define ROWL    (THREADS / BC)  // 128 row-lanes

// ---------------- LDS layout (floats), buf0 reused for L0 then L2 ----------
#define ENC_PAD  2052
#define BUF0_OFF (ENC_PAD * BC)            //  8208
#define BUF1_OFF (BUF0_OFF + N0 * BC)      // 40976
#define SMEM_F   (BUF1_OFF + N1 * BC)      // 57360 floats = 229,440 B < 320 KB

// ---------------- CDNA5 async global->LDS staging ---------------------------
typedef int v4i __attribute__((vector_size(16)));

#if __has_builtin(__builtin_amdgcn_global_load_async_to_lds_b128)
#define HAVE_ASYNC_LDS 1
#else
#define HAVE_ASYNC_LDS 0
#endif

__device__ __forceinline__ void wait_async_zero() {
#if __has_builtin(__builtin_amdgcn_s_wait_asynccnt)
  __builtin_amdgcn_s_wait_asynccnt(0);
#else
  asm volatile("s_wait_asynccnt 0" ::: "memory");
#endif
}

// log(1 - exp(x)) for x < 0 (Maechler 2012), matches reference branch point.
__device__ __forceinline__ float log1mexp_f(float x) {
  if (x > -0.6931471805599453f) return __logf(-expm1f(x));
  return log1pf(-__expf(x));
}

__device__ __forceinline__ float lse8(float v0, float v1, float v2, float v3,
                                      float v4, float v5, float v6, float v7) {
  float m = fmaxf(fmaxf(fmaxf(v0, v1), fmaxf(v2, v3)),
                  fmaxf(fmaxf(v4, v5), fmaxf(v6, v7)));
  if (!(m > -__builtin_inff())) return m;   // all -inf (defensive; can't occur here)
  float s = __expf(v0 - m) + __expf(v1 - m) + __expf(v2 - m) + __expf(v3 - m)
          + __expf(v4 - m) + __expf(v5 - m) + __expf(v6 - m) + __expf(v7 - m);
  return m + __logf(s);
}

__global__ __launch_bounds__(THREADS, 1)
void kl_fused(const float* __restrict__ pos,
              const int*   __restrict__ idx0,
              const int*   __restrict__ idx1,
              const int*   __restrict__ idx2,
              const int*   __restrict__ idx3,
              float*       __restrict__ out) {
  __shared__ float smem[SMEM_F];
  float* enc  = smem;             // [ENC_PAD][BC]
  float* buf0 = smem + BUF0_OFF;  // [N0][BC]   (L0 result, later L2 result)
  float* buf1 = smem + BUF1_OFF;  // [N1][BC]   (L1 result)

  const int t       = threadIdx.x;
  const int c       = t & (BC - 1);
  const int rl      = t >> 2;                 // 0..ROWL-1
  const int colBase = blockIdx.x * BC;

  // ---- Stage pos tile into the even (pos) rows of enc -----------------------
#if HAVE_ASYNC_LDS
  for (int v = t; v < NVARS; v += THREADS) {
    __builtin_amdgcn_global_load_async_to_lds_b128(
        (v4i*)(pos + (size_t)v * BATCH + colBase),   // global src (16B per lane)
        (v4i*)(enc + (2 + 2 * v) * BC),              // LDS dst (even enc row)
        0, 0);
  }
  wait_async_zero();
#else
  for (int v = t; v < NVARS; v += THREADS) {
    float4 p = *(const float4*)(pos + (size_t)v * BATCH + colBase);
    *(float4*)(enc + (2 + 2 * v) * BC) = p;
  }
#endif
  if (t < BC) { enc[0 * BC + t] = -__builtin_inff(); enc[1 * BC + t] = 0.0f; }
  __syncthreads();

  // ---- Fill odd (neg) rows: neg = log1mexp(pos) -----------------------------
  for (int v = t; v < NVARS; v += THREADS) {
    float4 p = *(const float4*)(enc + (2 + 2 * v) * BC);
    float4 n;
    n.x = log1mexp_f(p.x); n.y = log1mexp_f(p.y);
    n.z = log1mexp_f(p.z); n.w = log1mexp_f(p.w);
    *(float4*)(enc + (3 + 2 * v) * BC) = n;
  }
  __syncthreads();

  // ---- L0: product layer (gather 4 + sum) -> buf0 ---------------------------
  {
    const int4* id = (const int4*)idx0;
    for (int o = rl; o < N0; o += ROWL) {
      int4 ix = id[o];
      buf0[o * BC + c] = enc[ix.x * BC + c] + enc[ix.y * BC + c]
                       + enc[ix.z * BC + c] + enc[ix.w * BC + c];
    }
  }
  __syncthreads();

  // ---- L1: sum layer (gather 8 + logsumexp) -> buf1 -------------------------
  for (int s = rl; s < N1; s += ROWL) {
    const int4* p8 = (const int4*)(idx1 + s * 8);
    int4 a = p8[0], b = p8[1];
    buf1[s * BC + c] = lse8(buf0[a.x * BC + c], buf0[a.y * BC + c],
                            buf0[a.z * BC + c], buf0[a.w * BC + c],
                            buf0[b.x * BC + c], buf0[b.y * BC + c],
                            buf0[b.z * BC + c], buf0[b.w * BC + c]);
  }
  __syncthreads();

  // ---- L2: product layer -> buf0 (overwrites dead L0 data) ------------------
  {
    const int4* id = (const int4*)idx2;
    for (int o = rl; o < N2; o += ROWL) {
      int4 ix = id[o];
      buf0[o * BC + c] = buf1[ix.x * BC + c] + buf1[ix.y * BC + c]
                       + buf1[ix.z * BC + c] + buf1[ix.w * BC + c];
    }
  }
  __syncthreads();

  // ---- L3: sum layer -> global output ---------------------------------------
  for (int s = rl; s < N3; s += ROWL) {
    const int4* p8 = (const int4*)(idx3 + s * 8);
    int4 a = p8[0], b = p8[1];
    float r = lse8(buf0[a.x * BC + c], buf0[a.y * BC + c],
                   buf0[a.z * BC + c], buf0[a.w * BC + c],
                   buf0[b.x * BC + c], buf0[b.y * BC + c],
                   buf0[b.z * BC + c], buf0[b.w * BC + c]);
    out[(size_t)s * BATCH + colBase + c] = r;
  }
}

extern "C" void kernel_launch(void* const* d_in, const int* in_sizes, int n_in,
                              void* d_out, int out_size, void* d_ws, size_t ws_size,
                              hipStream_t stream) {
  (void)in_sizes; (void)n_in; (void)out_size; (void)d_ws; (void)ws_size;
  const float* pos  = (const float*)d_in[0];
  const int*   idx0 = (const int*)d_in[1];
  const int*   idx1 = (const int*)d_in[2];
  const int*   idx2 = (const int*)d_in[3];
  const int*   idx3 = (const int*)d_in[4];
  float* out = (float*)d_out;

  dim3 grid(BATCH / BC);   // 1024 workgroups, one 224KB LDS tile each
  hipLaunchKernelGGL(kl_fused, grid, dim3(THREADS), 0, stream,
                     pos, idx0, idx1, idx2, idx3, out);
}